// SmoothLDDTLoss_927712936252
// MI455X (gfx1250) — compile-verified
//
#include <hip/hip_runtime.h>

typedef __attribute__((ext_vector_type(2))) float v2f;
typedef __attribute__((ext_vector_type(8))) float v8f;

// ---------------------------------------------------------------------------
// Zero the per-batch accumulators (graph-capture safe, deterministic).
// ---------------------------------------------------------------------------
__global__ void lddt_init_acc(float* __restrict__ acc, int nacc) {
    int i = blockIdx.x * blockDim.x + threadIdx.x;
    if (i < nacc) acc[i] = 0.0f;
}

// ---------------------------------------------------------------------------
// Per-point precompute: squared norms of pred/true coords + packed flags.
// flags bit0 = is_nucleotide (dna|rna), bit1 = coords_mask.
// ---------------------------------------------------------------------------
__global__ void lddt_precompute(const float* __restrict__ pred,
                                const float* __restrict__ truec,
                                const int* __restrict__ is_dna,
                                const int* __restrict__ is_rna,
                                const int* __restrict__ cmask,
                                float* __restrict__ sqp,
                                float* __restrict__ sqt,
                                int* __restrict__ flags,
                                int total) {
    int i = blockIdx.x * blockDim.x + threadIdx.x;
    if (i >= total) return;
    const float* p = pred  + (size_t)i * 3;
    const float* t = truec + (size_t)i * 3;
    sqp[i] = p[0] * p[0] + p[1] * p[1] + p[2] * p[2];
    sqt[i] = t[0] * t[0] + t[1] * t[1] + t[2] * t[2];
    int nuc = ((is_dna[i] != 0) || (is_rna[i] != 0)) ? 1 : 0;
    int cm  = (cmask[i] != 0) ? 2 : 0;
    flags[i] = nuc | cm;
}

// ---------------------------------------------------------------------------
// Main: symmetry-halved tiling. The pair matrix is symmetric (eps and mask
// both symmetric), so sum(full) = sum(diag tiles) + 2 * sum(upper tiles).
// Round-robin cover: wave (I, J') computes tile (I, (I+J') mod T), J' in
// [0, T/2]; weight 1 for J'==0 (diagonal) and J'==T/2 (each unordered pair
// appears twice as transposes), weight 2 otherwise. Every wave does a full
// valid tile -> EXEC is all-ones for both WMMAs, no ragged waves.
//
// Each wave32 owns one 16x16 tile; Gram inner products via
// V_WMMA_F32_16X16X4_F32 (K=3 padded to 4, exact f32 -> matches reference).
// Block = 256 threads = 8 waves stacked along I.
// ---------------------------------------------------------------------------
__global__ void __launch_bounds__(256)
lddt_main(const float* __restrict__ pred,
          const float* __restrict__ truec,
          const float* __restrict__ sqp,
          const float* __restrict__ sqt,
          const int*   __restrict__ flags,
          float* __restrict__ acc,
          int n) {
    const int T     = n >> 4;                     // #16-tiles per side (256)
    const int lane  = threadIdx.x & 31;
    const int wave  = threadIdx.x >> 5;
    const int batch = blockIdx.z;
    const int I     = blockIdx.y * 8 + wave;      // row tile (8 per block)
    const int Jp    = blockIdx.x;                 // J' in [0, T/2]
    const int Jw    = (I + Jp) & (T - 1);         // wrapped column tile
    const int m     = lane & 15;
    const int hi    = lane >> 4;                  // lane half selects K pair
    const size_t base = (size_t)batch * (size_t)n;

    const float weight = (Jp == 0 || Jp == (T >> 1)) ? 1.0f : 2.0f;

    const int gi = I  * 16 + m;                   // row index this lane loads
    const int gj = Jw * 16 + m;                   // col index this lane loads

    // Load full xyz for the row/col points (b96-friendly), select K-half
    // branchlessly so EXEC stays all-ones for the WMMAs.
    const float* pr = pred  + (base + (size_t)gi) * 3;
    const float* pc = pred  + (base + (size_t)gj) * 3;
    const float* tr = truec + (base + (size_t)gi) * 3;
    const float* tc = truec + (base + (size_t)gj) * 3;

    float prx = pr[0], pry = pr[1], prz = pr[2];
    float pcx = pc[0], pcy = pc[1], pcz = pc[2];
    float trx = tr[0], try_ = tr[1], trz = tr[2];
    float tcx = tc[0], tcy = tc[1], tcz = tc[2];

    v2f ap, bp, at, bt;
    ap.x = hi ? prz : prx;  ap.y = hi ? 0.0f : pry;   // A: rows, K0/K1 | K2/K3
    bp.x = hi ? pcz : pcx;  bp.y = hi ? 0.0f : pcy;   // B: cols of X^T
    at.x = hi ? trz : trx;  at.y = hi ? 0.0f : try_;
    bt.x = hi ? tcz : tcx;  bt.y = hi ? 0.0f : tcy;

    v8f cz = {0.f, 0.f, 0.f, 0.f, 0.f, 0.f, 0.f, 0.f};
    // D[v] = Gram[row = v + 8*hi][col = lane&15] for this 16x16 tile.
    v8f gp = __builtin_amdgcn_wmma_f32_16x16x4_f32(false, ap, false, bp,
                                                   (short)0, cz, false, false);
    v8f gt = __builtin_amdgcn_wmma_f32_16x16x4_f32(false, at, false, bt,
                                                   (short)0, cz, false, false);

    // Column-side scalars for this lane (elementwise col index == gj).
    const float sqp_c = sqp[base + gj];
    const float sqt_c = sqt[base + gj];
    const int   fc    = flags[base + gj];

    float num = 0.0f, den = 0.0f;
#pragma unroll
    for (int v = 0; v < 8; ++v) {
        const int r = I * 16 + v + 8 * hi;        // global row of D element v
        const float d2p = sqp[base + r] + sqp_c - 2.0f * gp[v];
        const float d2t = sqt[base + r] + sqt_c - 2.0f * gt[v];
        const float dp = __builtin_amdgcn_sqrtf(fmaxf(d2p, 0.0f));
        const float dt = __builtin_amdgcn_sqrtf(fmaxf(d2t, 0.0f));
        const float diff = fabsf(dt - dp);

        // sum_i sigmoid(t_i - diff) = sum_i 1/(1 + e^{diff} * e^{-t_i})
        const float e = __builtin_amdgcn_exp2f(diff * 1.4426950408889634f);
        float s = __builtin_amdgcn_rcpf(1.0f + e * 0.60653065971263342f)   // 0.5
                + __builtin_amdgcn_rcpf(1.0f + e * 0.36787944117144233f)   // 1
                + __builtin_amdgcn_rcpf(1.0f + e * 0.13533528323661270f)   // 2
                + __builtin_amdgcn_rcpf(1.0f + e * 0.01831563888873418f);  // 4

        const int fr = flags[base + r];
        const int fb = fr & fc;
        const float cutoff = (fb & 1) ? 30.0f : 15.0f;      // nucleotide pair?
        const bool ok = (dt < cutoff) && (r != gj) && ((fb & 2) != 0);
        num += ok ? 0.25f * s : 0.0f;
        den += ok ? 1.0f : 0.0f;
    }

    // Apply symmetry weight, then wave32 -> block reduction -> 2 atomics.
    num *= weight;
    den *= weight;
#pragma unroll
    for (int off = 16; off > 0; off >>= 1) {
        num += __shfl_xor(num, off, 32);
        den += __shfl_xor(den, off, 32);
    }
    __shared__ float snum[8];
    __shared__ float sden[8];
    if (lane == 0) { snum[wave] = num; sden[wave] = den; }
    __syncthreads();
    if (threadIdx.x == 0) {
        float tn = 0.0f, td = 0.0f;
#pragma unroll
        for (int w = 0; w < 8; ++w) { tn += snum[w]; td += sden[w]; }
        atomicAdd(&acc[batch * 2 + 0], tn);
        atomicAdd(&acc[batch * 2 + 1], td);
    }
}

// ---------------------------------------------------------------------------
// Finalize: out = 1 - mean_b( num_b / max(den_b, 1) )
// ---------------------------------------------------------------------------
__global__ void lddt_finalize(const float* __restrict__ acc,
                              float* __restrict__ out, int b) {
    if (blockIdx.x == 0 && threadIdx.x == 0) {
        float s = 0.0f;
        for (int i = 0; i < b; ++i) {
            const float num = acc[i * 2 + 0];
            const float den = fmaxf(acc[i * 2 + 1], 1.0f);
            s += num / den;
        }
        out[0] = 1.0f - s / (float)b;
    }
}

// ---------------------------------------------------------------------------
extern "C" void kernel_launch(void* const* d_in, const int* in_sizes, int n_in,
                              void* d_out, int out_size, void* d_ws, size_t ws_size,
                              hipStream_t stream) {
    const float* pred   = (const float*)d_in[0];
    const float* truec  = (const float*)d_in[1];
    const int*   is_dna = (const int*)d_in[2];
    const int*   is_rna = (const int*)d_in[3];
    const int*   cmask  = (const int*)d_in[4];
    float* out = (float*)d_out;

    const int n     = 4096;
    const int total = in_sizes[2];        // b * n
    const int b     = total / n;
    const int T     = n >> 4;             // 256 tiles per side

    // Workspace layout (floats): [0..15] accumulators (num/den per batch),
    // then sq_pred[total], sq_true[total], flags[total] (int).
    float* acc   = (float*)d_ws;
    float* sqp   = acc + 16;
    float* sqt   = sqp + total;
    int*   flags = (int*)(sqt + total);

    lddt_init_acc<<<1, 32, 0, stream>>>(acc, 2 * b);
    lddt_precompute<<<(total + 255) / 256, 256, 0, stream>>>(
        pred, truec, is_dna, is_rna, cmask, sqp, sqt, flags, total);

    // J' in [0, T/2] (129 values), 8 row-tiles per block along I.
    dim3 grid(T / 2 + 1, T / 8, b);
    lddt_main<<<grid, 256, 0, stream>>>(pred, truec, sqp, sqt, flags, acc, n);

    lddt_finalize<<<1, 1, 0, stream>>>(acc, out, b);
}